// IF_31636729102644
// MI455X (gfx1250) — compile-verified
//
#include <hip/hip_runtime.h>
#include <cstdint>

// ---------------- problem constants (reference: T=8, B=32, C=128, H=W=32) ----
#define TT      8
#define BBATCH  32
#define CCHAN   128
#define HWPLANE 1024                  // 32*32
#define NSITE   (BBATCH*CCHAN*HWPLANE)   // 4,194,304 sites
#define PLANES  (BBATCH*CCHAN)           // 4096 (b,c) planes
#define CHUNKS  (NSITE/4)                // 1,048,576 (power of 2 = 2^20)

typedef __attribute__((ext_vector_type(2))) float v2f;
typedef __attribute__((ext_vector_type(8))) float v8f;

// Generic pointer to a __shared__ object: low 32 bits of the flat address are
// the LDS offset (flat-shared = {SHARED_BASE aperture, 32-bit LDS offset}).
__device__ __forceinline__ unsigned lds_off_of(const void* p) {
  return (unsigned)(unsigned long long)(uintptr_t)p;
}

// ---------------------------------------------------------------------------
__global__ void k0_zero(float* __restrict__ chan_sum) {
  chan_sum[threadIdx.x] = 0.0f;      // 128 threads
}

// ---------------------------------------------------------------------------
// One block per (b,c) plane of 1024 sites. Stages all 8 temporal slices of the
// plane (32 KB) into LDS via CDNA5 async-tensor DMA, runs the IF recurrence,
// emits packed spike bitmasks, and reduces new_thre into chan_sum[c].
__global__ __launch_bounds__(256) void k1_if_recurrence(
    const float* __restrict__ x, const float* __restrict__ thresh,
    unsigned* __restrict__ mask32, float* __restrict__ chan_sum) {
  __shared__ float sh[TT * HWPLANE];   // 32 KB: sh[t][site]
  __shared__ float wsum[8];

  const int    tid   = threadIdx.x;
  const int    plane = blockIdx.x;           // = b*C + c
  const int    chan  = plane & (CCHAN - 1);
  const size_t base  = (size_t)plane * HWPLANE;

  const float thre = thresh[0];

  // --- async global->LDS staging: one b128 per (thread, t) -----------------
  {
    const unsigned l0 = lds_off_of(&sh[0]) + (unsigned)tid * 16u;
    const float*   g0 = x + base + (size_t)tid * 4;
#pragma unroll
    for (int t = 0; t < TT; ++t) {
      unsigned           loff = l0 + (unsigned)t * (HWPLANE * 4);
      unsigned long long ga   = (unsigned long long)(uintptr_t)(g0 + (size_t)t * NSITE);
      asm volatile("global_load_async_to_lds_b128 %0, %1, off"
                   :: "v"(loff), "v"(ga) : "memory");
    }
  }
  asm volatile("s_wait_asynccnt 0" ::: "memory");
  // each thread consumes exactly the bytes it issued -> no workgroup barrier

  // --- IF recurrence over T for 4 consecutive sites ------------------------
  const float half_thre = 0.5f * thre;
  float    nt0 = 0.0f, nt1 = 0.0f;   // two partials (feeds the WMMA B operand)
  unsigned packed = 0u;
#pragma unroll
  for (int j = 0; j < 4; ++j) {
    const int site = tid * 4 + j;
    float    mem  = half_thre;
    float    cnt  = 0.0f;
    unsigned bits = 0u;
#pragma unroll
    for (int t = 0; t < TT; ++t) {
      mem += sh[t * HWPLANE + site];
      const bool  fire = (mem >= thre);        // heaviside(mem - thre)
      const float s    = fire ? 1.0f : 0.0f;
      bits |= fire ? (1u << t) : 0u;
      cnt  += s;
      mem  -= s * thre;                        // mem - s*cur
    }
    const float compen = (mem - half_thre) + cnt * thre;
    const float cv     = fminf(compen, (float)TT * thre);
    const float ntv    = (cv > 0.0f && cnt > 0.0f) ? (cv / cnt) : 0.0f;
    if (j < 2) nt0 += ntv; else nt1 += ntv;
    packed |= bits << (8 * j);
  }
  mask32[base / 4 + tid] = packed;   // 4 spike bytes in one b32 store

  // --- per-wave sum of new_thre --------------------------------------------
  float wave_total;
#if __has_builtin(__builtin_amdgcn_wmma_f32_16x16x4_f32)
  {
    // D = ones(16x4) * B(4x16): every row of D holds the exact fp32 column
    // sums of the 64 lane-partials; the grand total is layout-invariant.
    v2f a; a[0] = 1.0f; a[1] = 1.0f;
    v2f b; b[0] = nt0;  b[1] = nt1;
    v8f c = {};
    c = __builtin_amdgcn_wmma_f32_16x16x4_f32(false, a, false, b,
                                              (short)0, c, false, false);
    float cs = c[0];                  // colsum[n], duplicated in lanes 16-31
    cs += __shfl_xor(cs, 16);
    cs += __shfl_xor(cs, 8);
    cs += __shfl_xor(cs, 4);
    cs += __shfl_xor(cs, 2);
    cs += __shfl_xor(cs, 1);
    wave_total = 0.5f * cs;           // each column counted twice -> halve
  }
#else
  {
    float s = nt0 + nt1;
    s += __shfl_xor(s, 16);
    s += __shfl_xor(s, 8);
    s += __shfl_xor(s, 4);
    s += __shfl_xor(s, 2);
    s += __shfl_xor(s, 1);
    wave_total = s;
  }
#endif
  const int lane = tid & 31, wave = tid >> 5;
  if (lane == 0) wsum[wave] = wave_total;
  __syncthreads();
  if (tid == 0) {
    float s = 0.0f;
#pragma unroll
    for (int w = 0; w < 8; ++w) s += wsum[w];
    atomicAdd(&chan_sum[chan], s);    // global_atomic_add_f32
  }
}

// ---------------------------------------------------------------------------
// scale = thre + 0.2 * mean_c( (ub[c]-thre) * (thre > ub[c]) ),
// ub[c] = chan_sum[c] / (B*H*W = 32768)
__global__ void k2_update(const float* __restrict__ thresh,
                          const float* __restrict__ chan_sum,
                          float* __restrict__ scale) {
  __shared__ float w[4];
  const int   tid  = threadIdx.x;         // 0..127 = channel
  const float thre = thresh[0];
  const float ub   = chan_sum[tid] * (1.0f / 32768.0f);
  const float d    = ub - thre;
  float contrib    = (thre > ub) ? d : 0.0f;
  contrib += __shfl_xor(contrib, 16);
  contrib += __shfl_xor(contrib, 8);
  contrib += __shfl_xor(contrib, 4);
  contrib += __shfl_xor(contrib, 2);
  contrib += __shfl_xor(contrib, 1);
  if ((tid & 31) == 0) w[tid >> 5] = contrib;
  __syncthreads();
  if (tid == 0) {
    const float s      = w[0] + w[1] + w[2] + w[3];
    const float update = 0.1f * 2.0f * (s * (1.0f / (float)CCHAN));
    scale[0] = thre + update;
  }
}

// ---------------------------------------------------------------------------
// out[t, site] = spike(t, site) ? scale : 0 ; float4 stores, mask is L2-hot.
__global__ __launch_bounds__(256) void k3_scale(
    const unsigned* __restrict__ mask32, const float* __restrict__ scale_p,
    float4* __restrict__ out) {
  const int      idx   = blockIdx.x * 256 + threadIdx.x;  // 0 .. T*CHUNKS-1
  const int      t     = idx >> 20;                       // / CHUNKS (2^20)
  const int      chunk = idx & (CHUNKS - 1);
  const float    sc    = scale_p[0];
  const unsigned m4    = mask32[chunk];
  float4 o;
  o.x = ((m4 >> (t      )) & 1u) ? sc : 0.0f;
  o.y = ((m4 >> (t +  8 )) & 1u) ? sc : 0.0f;
  o.z = ((m4 >> (t + 16 )) & 1u) ? sc : 0.0f;
  o.w = ((m4 >> (t + 24 )) & 1u) ? sc : 0.0f;
  out[(size_t)t * CHUNKS + chunk] = o;
}

// ---------------------------------------------------------------------------
extern "C" void kernel_launch(void* const* d_in, const int* in_sizes, int n_in,
                              void* d_out, int out_size, void* d_ws, size_t ws_size,
                              hipStream_t stream) {
  const float* x      = (const float*)d_in[0];   // [T*B, C, H, W] fp32
  const float* thresh = (const float*)d_in[1];   // [1] fp32
  float*       out    = (float*)d_out;

  char*     ws       = (char*)d_ws;
  float*    chan_sum = (float*)(ws);             // 128 floats
  float*    scale    = (float*)(ws + 512);       // 1 float
  unsigned* mask32   = (unsigned*)(ws + 1024);   // NSITE bytes = 4 MB

  k0_zero<<<1, CCHAN, 0, stream>>>(chan_sum);
  k1_if_recurrence<<<PLANES, 256, 0, stream>>>(x, thresh, mask32, chan_sum);
  k2_update<<<1, CCHAN, 0, stream>>>(thresh, chan_sum, scale);
  k3_scale<<<(TT * CHUNKS) / 256, 256, 0, stream>>>(mask32, scale, (float4*)out);

  (void)in_sizes; (void)n_in; (void)out_size; (void)ws_size;
}